// RegionProposalNetwork_46677704573723
// MI455X (gfx1250) — compile-verified
//
#include <hip/hip_runtime.h>
#include <math.h>

typedef unsigned int u32;
typedef unsigned long long u64;
typedef float v2f __attribute__((ext_vector_type(2)));
typedef float v8f __attribute__((ext_vector_type(8)));

#define NIMG 8
#define NANCH 750000
#define PRE 4000
#define PAD 4096
#define POST 1000
#define IOU_T 0.7f
#define MINSZ 1e-3f
#define XCLIP 4.135166556742356f

// ---- workspace layout (u32 units) ----
#define OFF_HIST   0u          // 8 x 2048
#define OFF_MISC   16384u      // 8 x 16
#define OFF_SELKEY 16512u      // 8 x 4096
#define OFF_SELIDX 49280u      // 8 x 4096
#define OFF_BOXES  82048u      // 8 x 4096 x 4 floats (16B aligned)
#define OFF_AREA   213120u     // 8 x 4096 floats
#define OFF_PROB   245888u     // 8 x 4096 floats
#define OFF_VALID  278656u     // 8 x 4096 u32
#define OFF_KEEP   311424u     // 8 x 4096 u32
#define OFF_MASK   344192u     // 8 x 4096 x 256 u16  (4 MB u32)

__device__ __forceinline__ u32 f2key(float f) {
  u32 b = __float_as_uint(f);
  return b ^ ((b & 0x80000000u) ? 0xFFFFFFFFu : 0x80000000u);
}

__global__ void k_init(u32* hist, u32* misc) {
  int t = blockIdx.x * blockDim.x + threadIdx.x;
  for (int i = t; i < 16384; i += gridDim.x * blockDim.x) hist[i] = 0u;
  if (t < 128) misc[t] = ((t & 15) == 0) ? (u32)PRE : 0u;
}

__global__ void k_hist(const float* __restrict__ obj, u32* __restrict__ hist,
                       const u32* __restrict__ misc, int level) {
  int img = blockIdx.y;
  __shared__ u32 lh[2048];
  int t = threadIdx.x; // 256
  for (int i = t; i < 2048; i += 256) lh[i] = 0u;
  __syncthreads();
  u32 T1 = 0, T2 = 0;
  if (level >= 2) T1 = misc[img * 16 + 1];
  if (level >= 3) T2 = misc[img * 16 + 3];
  const float* o = obj + (size_t)img * NANCH;
  int stride = gridDim.x * 256;
  for (int i = blockIdx.x * 256 + t; i < NANCH; i += stride) {
    __builtin_prefetch(&o[i + stride], 0, 0);
    u32 key = f2key(o[i]);
    u32 bin;
    if (level == 1) bin = key >> 21;
    else if (level == 2) { if ((key >> 21) != T1) continue; bin = (key >> 10) & 0x7FFu; }
    else { if ((key >> 21) != T1 || (((key >> 10) & 0x7FFu) != T2)) continue; bin = key & 0x3FFu; }
    atomicAdd(&lh[bin], 1u);
  }
  __syncthreads();
  u32* gh = hist + img * 2048;
  for (int i = t; i < 2048; i += 256) if (lh[i]) atomicAdd(&gh[i], lh[i]);
}

__global__ void k_find(u32* hist, u32* misc, int level) {
  int img = blockIdx.x;
  u32* gh = hist + img * 2048;
  u32* m  = misc + img * 16;
  if (threadIdx.x == 0) {
    u32 need = m[2 * (level - 1)];
    u32 cum = 0, T = 0, rem = need;
    for (int bin = 2047; bin >= 0; --bin) {
      u32 c = gh[bin];
      if (cum + c >= need) { T = (u32)bin; rem = need - cum; break; }
      cum += c;
    }
    m[2 * (level - 1) + 1] = T;
    m[2 * level] = rem;
  }
  __syncthreads();
  for (int i = threadIdx.x; i < 2048; i += 256) gh[i] = 0u; // ready for next level
}

__global__ void k_compact(const float* __restrict__ obj, u32* misc,
                          u32* __restrict__ selkey, u32* __restrict__ selidx) {
  int img = blockIdx.y;
  int t = threadIdx.x;
  u32* m = misc + img * 16;
  u32 T1 = m[1], T2 = m[3], T3 = m[5], quota = m[6];
  const float* o = obj + (size_t)img * NANCH;
  int stride = gridDim.x * 256;
  for (int i = blockIdx.x * 256 + t; i < NANCH; i += stride) {
    u32 key = f2key(o[i]);
    u32 b1 = key >> 21, b2 = (key >> 10) & 0x7FFu, b3 = key & 0x3FFu;
    bool acc = false;
    if (b1 > T1) acc = true;
    else if (b1 == T1) {
      if (b2 > T2) acc = true;
      else if (b2 == T2) {
        if (b3 > T3) acc = true;
        else if (b3 == T3) acc = (atomicAdd(&m[7], 1u) < quota);
      }
    }
    if (acc) {
      u32 slot = atomicAdd(&m[8], 1u);
      selkey[img * PAD + slot] = key;
      selidx[img * PAD + slot] = (u32)i;
    }
  }
}

// bitonic sort (descending) of 4096 (key<<32 | ~idx), then gather+decode boxes
__global__ void k_sortdecode(const u32* __restrict__ selkey, const u32* __restrict__ selidx,
                             const float* __restrict__ deltas, const float* __restrict__ anchors,
                             float* __restrict__ boxes, float* __restrict__ area,
                             float* __restrict__ prob, u32* __restrict__ validm) {
  int img = blockIdx.x;
  int t = threadIdx.x; // 1024
  __shared__ u64 sd[PAD];
  for (int s = t; s < PAD; s += 1024) {
    u64 v = 0ull;
    if (s < PRE) {
      u32 k = selkey[img * PAD + s];
      u32 id = selidx[img * PAD + s];
      v = ((u64)k << 32) | (u32)(~id);
    }
    sd[s] = v;
  }
  __syncthreads();
  for (int k = 2; k <= PAD; k <<= 1) {
    for (int j = k >> 1; j > 0; j >>= 1) {
      for (int s = t; s < PAD; s += 1024) {
        int p = s ^ j;
        if (p > s) {
          bool desc = ((s & k) == 0);
          u64 a0 = sd[s], a1 = sd[p];
          bool sw = desc ? (a0 < a1) : (a0 > a1);
          if (sw) { sd[s] = a1; sd[p] = a0; }
        }
      }
      __syncthreads();
    }
  }
  for (int s = t; s < PAD; s += 1024) {
    int gi = img * PAD + s;
    if (s < PRE) {
      u64 v = sd[s];
      u32 key = (u32)(v >> 32);
      u32 id  = ~((u32)v);
      u32 bits = (key & 0x80000000u) ? (key ^ 0x80000000u) : ~key;
      float sc = __uint_as_float(bits);
      float pr = 1.0f / (1.0f + expf(-sc));
      const float* d = deltas + ((size_t)img * NANCH + id) * 4;
      const float* a = anchors + (size_t)id * 4;
      float aw = a[2] - a[0], ah = a[3] - a[1];
      float cx = a[0] + 0.5f * aw, cy = a[1] + 0.5f * ah;
      float dx = d[0], dy = d[1];
      float dw = fminf(d[2], XCLIP), dh = fminf(d[3], XCLIP);
      float pcx = dx * aw + cx, pcy = dy * ah + cy;
      float pw = expf(dw) * aw, ph = expf(dh) * ah;
      float x0 = fminf(fmaxf(pcx - 0.5f * pw, 0.0f), 1000.0f);
      float y0 = fminf(fmaxf(pcy - 0.5f * ph, 0.0f), 1000.0f);
      float x1 = fminf(fmaxf(pcx + 0.5f * pw, 0.0f), 1000.0f);
      float y1 = fminf(fmaxf(pcy + 0.5f * ph, 0.0f), 1000.0f);
      bool vld = (x1 - x0 >= MINSZ) && (y1 - y0 >= MINSZ);
      float4 bx; bx.x = x0; bx.y = y0; bx.z = x1; bx.w = y1;
      ((float4*)boxes)[gi] = bx;
      area[gi] = (x1 - x0) * (y1 - y0);
      prob[gi] = pr;
      validm[gi] = vld ? 1u : 0u;
    } else {
      float4 z; z.x = 0.f; z.y = 0.f; z.z = 0.f; z.w = 0.f;
      ((float4*)boxes)[gi] = z;
      area[gi] = 0.0f;
      prob[gi] = -INFINITY;
      validm[gi] = 0u;
    }
  }
}

// One wave per 16x16 IoU tile. Union base term area_i + area_j computed as a
// rank-2 outer-sum via V_WMMA_F32_16X16X4_F32: A rows = [area_i, 1, 0, 0],
// B cols = [1, area_j, 0, 0]. Suppression bits packed with wave32 ballot.
__global__ void k_ioumask(const float* __restrict__ boxes, const float* __restrict__ area,
                          unsigned short* __restrict__ mask) {
  int img = blockIdx.y;
  int tile = blockIdx.x;           // 256*256 tiles
  int rt = tile >> 8, ct = tile & 255;
  int lane = threadIdx.x;          // 0..31
  int n = lane & 15;
  int half = lane >> 4;
  const float4* bb = ((const float4*)boxes) + (size_t)img * PAD;
  const float*  ab = area + (size_t)img * PAD;

  float arow = ab[rt * 16 + n];    // unconditional load, lane n -> row n of tile
  float acol = ab[ct * 16 + n];
  v2f A, Bm;
  A.x  = half ? 0.0f : arow;       // K=0: row areas   (K=2 half -> 0)
  A.y  = half ? 0.0f : 1.0f;       // K=1: ones        (K=3 half -> 0)
  Bm.x = half ? 0.0f : 1.0f;       // K=0: ones row
  Bm.y = half ? 0.0f : acol;       // K=1: col areas
  v8f C = {};
  C = __builtin_amdgcn_wmma_f32_16x16x4_f32(false, A, false, Bm, (short)0, C, false, false);

  float4 bj = bb[ct * 16 + n];
  u32 rowbase = (u32)(img * PAD + rt * 16);
  for (int r = 0; r < 8; ++r) {
    int m = r + 8 * half;
    float4 bi = bb[rt * 16 + m];
    float ltx = fmaxf(bi.x, bj.x), lty = fmaxf(bi.y, bj.y);
    float rbx = fminf(bi.z, bj.z), rby = fminf(bi.w, bj.w);
    float iw = fmaxf(rbx - ltx, 0.0f), ih = fmaxf(rby - lty, 0.0f);
    float inter = iw * ih;
    float uni = C[r] - inter;
    float iou = inter / fmaxf(uni, 1e-9f);
    u32 bal = __builtin_amdgcn_ballot_w32(iou > IOU_T);
    if (lane == 0)  mask[(rowbase + r)     * 256 + ct] = (unsigned short)(bal & 0xFFFFu);
    if (lane == 16) mask[(rowbase + r + 8) * 256 + ct] = (unsigned short)(bal >> 16);
  }
}

// sequential greedy NMS over precomputed suppression bitmask
__global__ void k_nms(const unsigned short* __restrict__ mask,
                      const u32* __restrict__ validm, u32* __restrict__ keep) {
  int img = blockIdx.x;
  int t = threadIdx.x; // 128 threads, each owns one 32-bit word of 4096 bits
  __shared__ u32 removed[128];
  __shared__ int sh_keep;
  const u32* v = validm + img * PAD;
  u32 w = 0;
  for (int b = 0; b < 32; ++b) w |= (v[t * 32 + b] ? 0u : 1u) << b;
  removed[t] = w;
  __syncthreads();
  const u32* mrow = ((const u32*)mask) + (size_t)img * PAD * 128;
  u32* kp = keep + img * PAD;
  for (int i = 0; i < PRE; ++i) {
    if (t == 0) {
      int k = (int)(((removed[i >> 5] >> (i & 31)) & 1u) ^ 1u);
      sh_keep = k;
      kp[i] = (u32)k;
    }
    __syncthreads();
    if (sh_keep) removed[t] |= mrow[(size_t)i * 128 + t];
    __syncthreads();
  }
}

// compaction: kept entries (already score-descending) first, then non-kept w/ -inf
__global__ void k_out(const float* __restrict__ boxes, const float* __restrict__ prob,
                      const u32* __restrict__ keep, float* __restrict__ out) {
  int img = blockIdx.x;
  int t = threadIdx.x; // 1024
  __shared__ u32 s[1024];
  int i0 = 4 * t;
  u32 k[4]; u32 cnt = 0;
  for (int u = 0; u < 4; ++u) {
    int i = i0 + u;
    k[u] = (i < PRE) ? keep[img * PAD + i] : 0u;
    cnt += k[u];
  }
  s[t] = cnt;
  for (int d = 1; d < 1024; d <<= 1) {
    __syncthreads();
    u32 val = (t >= d) ? s[t - d] : 0u;
    __syncthreads();
    s[t] += val;
  }
  __syncthreads();
  u32 total = s[1023];
  u32 kr = s[t] - cnt; // exclusive prefix of kept
  for (int u = 0; u < 4; ++u) {
    int i = i0 + u;
    if (i >= PRE) break;
    u32 slot = k[u] ? kr : (total + (u32)i - kr);
    if (slot < POST) {
      const float* b = boxes + ((size_t)img * PAD + i) * 4;
      float* ob = out + ((size_t)img * POST + slot) * 4;
      ob[0] = b[0]; ob[1] = b[1]; ob[2] = b[2]; ob[3] = b[3];
      out[(size_t)NIMG * POST * 4 + img * POST + slot] =
          k[u] ? prob[img * PAD + i] : -INFINITY;
    }
    kr += k[u];
  }
}

extern "C" void kernel_launch(void* const* d_in, const int* in_sizes, int n_in,
                              void* d_out, int out_size, void* d_ws, size_t ws_size,
                              hipStream_t stream) {
  const float* obj     = (const float*)d_in[0]; // (8, 750000)
  const float* deltas  = (const float*)d_in[1]; // (8, 750000, 4)
  const float* anchors = (const float*)d_in[2]; // (750000, 4)
  float* out = (float*)d_out;                   // 32000 proposals + 8000 scores

  u32* ws = (u32*)d_ws;
  u32* hist   = ws + OFF_HIST;
  u32* misc   = ws + OFF_MISC;
  u32* selkey = ws + OFF_SELKEY;
  u32* selidx = ws + OFF_SELIDX;
  float* boxes = (float*)(ws + OFF_BOXES);
  float* area  = (float*)(ws + OFF_AREA);
  float* prob  = (float*)(ws + OFF_PROB);
  u32* validm  = ws + OFF_VALID;
  u32* keep    = ws + OFF_KEEP;
  unsigned short* mask = (unsigned short*)(ws + OFF_MASK);

  k_init<<<64, 256, 0, stream>>>(hist, misc);

  dim3 hgrid(48, NIMG);
  k_hist<<<hgrid, 256, 0, stream>>>(obj, hist, misc, 1);
  k_find<<<NIMG, 256, 0, stream>>>(hist, misc, 1);
  k_hist<<<hgrid, 256, 0, stream>>>(obj, hist, misc, 2);
  k_find<<<NIMG, 256, 0, stream>>>(hist, misc, 2);
  k_hist<<<hgrid, 256, 0, stream>>>(obj, hist, misc, 3);
  k_find<<<NIMG, 256, 0, stream>>>(hist, misc, 3);

  k_compact<<<hgrid, 256, 0, stream>>>(obj, misc, selkey, selidx);
  k_sortdecode<<<NIMG, 1024, 0, stream>>>(selkey, selidx, deltas, anchors,
                                          boxes, area, prob, validm);
  k_ioumask<<<dim3(256 * 256, NIMG), 32, 0, stream>>>(boxes, area, mask);
  k_nms<<<NIMG, 128, 0, stream>>>(mask, validm, keep);
  k_out<<<NIMG, 1024, 0, stream>>>(boxes, prob, keep, out);
}